// SphericalHarmonicsShellsConv_65309272703471
// MI455X (gfx1250) — compile-verified
//
#include <hip/hip_runtime.h>
#include <cmath>
#include <complex>
#include <vector>
#include <algorithm>
#include <cstring>

// ---------------------------------------------------------------------------
// Geometry (fixed by the reference):
//   B=4, N=4096, V=2048, P=32, C=16
//   channels: x0:16, x1:48, x2:80, x3:112  -> CH=256, c_off={0,16,64,144}
//   SPLIT_SIZE={4,9,10,7} -> Y=30, y_off={0,4,13,23}
// Outputs: out0 (1 x 160), out1 (3 x 336), out2 (5 x 384), out3 (7 x 320)
// ---------------------------------------------------------------------------
#define TB 256
#define GSTR 260          // LDS row stride (260 mod 64 == 4 -> conflict-free)

constexpr int YOFFc[4] = {0, 4, 13, 23};
constexpr int COFFc[4] = {0, 16, 64, 144};

// per-J output widths and flat offsets in d_out (floats)
constexpr size_t OUT0 = 0;                         // 8192 * 1 * 160
constexpr size_t OUT1 = 1310720;                   // + 8192 * 3 * 336
constexpr size_t OUT2 = 9568256;                   // + 8192 * 5 * 384
constexpr size_t OUT3 = 25296896;                  // + 8192 * 7 * 320

typedef __attribute__((ext_vector_type(2))) float v2f;
typedef __attribute__((ext_vector_type(8))) float v8f;

// ---------------------------------------------------------------------------
// Stage-D helpers (all compile-time specialized: no runtime divides)
// ---------------------------------------------------------------------------
template<int j>
__device__ __forceinline__ void direct_l0(const float* __restrict__ Y,
                                          float* __restrict__ dst, int W, int dofs, int tid) {
  constexpr int rows = 2 * j + 1, width = (4 - j) * 16, stride = 4 - j, yo = YOFFc[j];
  for (int i = tid; i < rows * width; i += TB) {
    const int mj = i / width, t = i - mj * width;
    const int kj = t >> 4, ch = t & 15;
    dst[mj * W + dofs + t] = Y[(yo + mj * stride + kj) * GSTR + ch];
  }
}

template<int l>
__device__ __forceinline__ void direct_j0(const float* __restrict__ Y,
                                          float* __restrict__ dst, int W, int dofs, int tid) {
  constexpr int rows = 2 * l + 1, co = COFFc[l];
  for (int i = tid; i < rows * 64; i += TB) {
    const int ml = i >> 6, t = i & 63;
    const int kj = t >> 4, ch = t & 15;
    dst[ml * W + dofs + t] = Y[kj * GSTR + co + ml * 16 + ch];
  }
}

template<int j, int l, int J>
__device__ __forceinline__ void cg_seg(const float* __restrict__ cg, int cgofs,
                                       const float* __restrict__ Y,
                                       float* __restrict__ dst, int W, int dofs, int tid) {
  constexpr int rows = 2 * J + 1, width = (4 - j) * 16;
  constexpr int nj = 2 * j + 1, nl = 2 * l + 1;
  constexpr int yo = YOFFc[j], co = COFFc[l], stride = 4 - j;
  for (int i = tid; i < rows * width; i += TB) {
    const int mJ = i / width, t = i - mJ * width;
    const int kj = t >> 4, ch = t & 15;
    float s = 0.f;
#pragma unroll
    for (int mj = 0; mj < nj; ++mj)
#pragma unroll
      for (int ml = 0; ml < nl; ++ml)
        s += cg[cgofs + (mj * nl + ml) * rows + mJ] *
             Y[(yo + mj * stride + kj) * GSTR + co + ml * 16 + ch];
    dst[mJ * W + dofs + t] = s;
  }
}

// ---------------------------------------------------------------------------
// Fused kernel: gather -> WMMA f32 GEMM (30x32 @ 32x256) -> CG recoupling
// One workgroup per (b,v); 256 threads = 8 wave32s.
// ---------------------------------------------------------------------------
__global__ __launch_bounds__(TB)
void shconv_kernel(const float* __restrict__ x0, const float* __restrict__ x1,
                   const float* __restrict__ x2, const float* __restrict__ x3,
                   const int* __restrict__ pidx, const float* __restrict__ kern,
                   const float* __restrict__ cg, float* __restrict__ out) {
  __shared__ float lds_k[32 * 32];       // A: kernels^T, [p][yy], yy padded to 32
  __shared__ float lds_gy[32 * GSTR];    // g (32x256) then reused as y (30x256)
  __shared__ int   lds_row[32];          // gathered row index b*N + n

  const int tid = threadIdx.x;
  const size_t bid = blockIdx.x;         // b*V + v

  // ---- Stage A: indices + kernel tile + gather -------------------------
  if (tid < 32) {
    const int* pp = pidx + (bid * 32 + (size_t)tid) * 2;
    lds_row[tid] = pp[0] * 4096 + pp[1];
  }
  for (int i = tid; i < 32 * 30; i += TB)
    lds_k[(i / 30) * 32 + (i % 30)] = kern[bid * (32 * 30) + i];
  for (int i = tid; i < 64; i += TB)     // zero pad rows yy = 30, 31
    lds_k[(i >> 1) * 32 + 30 + (i & 1)] = 0.f;
  __syncthreads();

#pragma unroll 4
  for (int p = 0; p < 32; ++p) {         // tid == channel (0..255), coalesced
    const int row = lds_row[p];
    const int c = tid;
    float v;
    if (c < 16)       v = x0[(size_t)row * 16  + c];
    else if (c < 64)  v = x1[(size_t)row * 48  + (c - 16)];
    else if (c < 144) v = x2[(size_t)row * 80  + (c - 64)];
    else              v = x3[(size_t)row * 112 + (c - 144)];
    lds_gy[p * GSTR + c] = v;
  }
  __syncthreads();

  // ---- Stage B: y = K^T * G via V_WMMA_F32_16X16X4_F32 ------------------
  const int wave = tid >> 5, lane = tid & 31;
  const int half = lane >> 4, l16 = lane & 15;
  v8f acc[4];

#pragma unroll
  for (int q = 0; q < 4; ++q) {
    const int tt = wave * 4 + q;
    const int nt = tt >> 1, mt = tt & 1;
    const int m = mt * 16 + l16;         // A row (yy), rows 30/31 discarded
    const int n = nt * 16 + l16;         // B col (channel)
    v8f c = {};
#pragma unroll
    for (int kk = 0; kk < 8; ++kk) {
      const int p0 = kk * 4 + half * 2;  // K pair per lane-half (ISA layout)
      v2f a, bf;
      a[0]  = lds_k[p0 * 32 + m];
      a[1]  = lds_k[(p0 + 1) * 32 + m];
      bf[0] = lds_gy[p0 * GSTR + n];
      bf[1] = lds_gy[(p0 + 1) * GSTR + n];
      c = __builtin_amdgcn_wmma_f32_16x16x4_f32(false, a, false, bf,
                                                (short)0, c, false, false);
    }
    acc[q] = c;
  }
  __syncthreads();                       // everyone done reading g

  // ---- Stage C: spill y (30x256) over the g buffer ----------------------
#pragma unroll
  for (int q = 0; q < 4; ++q) {
    const int tt = wave * 4 + q;
    const int nt = tt >> 1, mt = tt & 1;
#pragma unroll
    for (int r = 0; r < 8; ++r) {
      const int yy = mt * 16 + r + 8 * half;   // C/D layout: row = r + 8*half
      if (yy < 30) lds_gy[yy * GSTR + nt * 16 + l16] = acc[q][r];
    }
  }
  __syncthreads();

  // ---- Stage D: direct copies + CG contractions -------------------------
  const float* Y = lds_gy;
  float* o0 = out + OUT0 + bid * 160;
  float* o1 = out + OUT1 + bid * (3 * 336);
  float* o2 = out + OUT2 + bid * (5 * 384);
  float* o3 = out + OUT3 + bid * (7 * 320);

  // out0 (J=0, W=160)
  direct_l0<0>(Y, o0, 160, 0, tid);
  cg_seg<1,1,0>(cg,    0, Y, o0, 160,  64, tid);
  cg_seg<2,2,0>(cg,  783, Y, o0, 160, 112, tid);
  cg_seg<3,3,0>(cg, 2485, Y, o0, 160, 144, tid);
  // out1 (J=1, W=336)
  direct_l0<1>(Y, o1, 336, 0, tid);
  direct_j0<1>(Y, o1, 336, 48, tid);
  cg_seg<1,1,1>(cg,    9, Y, o1, 336, 112, tid);
  cg_seg<2,1,1>(cg,   81, Y, o1, 336, 160, tid);
  cg_seg<1,2,1>(cg,  558, Y, o1, 336, 192, tid);
  cg_seg<2,2,1>(cg,  808, Y, o1, 336, 240, tid);
  cg_seg<3,2,1>(cg, 1183, Y, o1, 336, 272, tid);
  cg_seg<2,3,1>(cg, 1960, Y, o1, 336, 288, tid);
  cg_seg<3,3,1>(cg, 2534, Y, o1, 336, 320, tid);
  // out2 (J=2, W=384)
  direct_l0<2>(Y, o2, 384, 0, tid);
  direct_j0<2>(Y, o2, 384, 32, tid);
  cg_seg<1,1,2>(cg,   36, Y, o2, 384,  96, tid);
  cg_seg<2,1,2>(cg,  126, Y, o2, 384, 144, tid);
  cg_seg<3,1,2>(cg,  306, Y, o2, 384, 176, tid);
  cg_seg<1,2,2>(cg,  603, Y, o2, 384, 192, tid);
  cg_seg<2,2,2>(cg,  883, Y, o2, 384, 240, tid);
  cg_seg<3,2,2>(cg, 1288, Y, o2, 384, 272, tid);
  cg_seg<1,3,2>(cg, 1708, Y, o2, 384, 288, tid);
  cg_seg<2,3,2>(cg, 2065, Y, o2, 384, 336, tid);
  cg_seg<3,3,2>(cg, 2681, Y, o2, 384, 368, tid);
  // out3 (J=3, W=320)
  direct_l0<3>(Y, o3, 320, 0, tid);
  direct_j0<3>(Y, o3, 320, 16, tid);
  cg_seg<2,1,3>(cg,  201, Y, o3, 320,  80, tid);
  cg_seg<3,1,3>(cg,  411, Y, o3, 320, 112, tid);
  cg_seg<1,2,3>(cg,  678, Y, o3, 320, 128, tid);
  cg_seg<2,2,3>(cg, 1008, Y, o3, 320, 176, tid);
  cg_seg<3,2,3>(cg, 1463, Y, o3, 320, 208, tid);
  cg_seg<1,3,3>(cg, 1813, Y, o3, 320, 224, tid);
  cg_seg<2,3,3>(cg, 2240, Y, o3, 320, 272, tid);
  cg_seg<3,3,3>(cg, 2926, Y, o3, 320, 304, tid);
}

// ---------------------------------------------------------------------------
// Host-side Clebsch-Gordan table (same math as the reference, double precision)
// ---------------------------------------------------------------------------
namespace cgbuild {
typedef std::complex<double> cd;

static double fct(int n) { double r = 1.0; for (int i = 2; i <= n; ++i) r *= (double)i; return r; }

static double su2_cg(int j1, int m1, int j2, int m2, int j3, int m3) {
  if (m3 != m1 + m2) return 0.0;
  const int vmin = std::max(std::max(-j1 + j2 + m3, -j1 + m1), 0);
  const int vmax = std::min(std::min(j2 + j3 + m1, j3 - j1 + j2), j3 + m3);
  const double C = std::sqrt((2.0 * j3 + 1.0) * fct(j3 + j1 - j2) * fct(j3 - j1 + j2) *
                             fct(j1 + j2 - j3) * fct(j3 + m3) * fct(j3 - m3) /
                             (fct(j1 + j2 + j3 + 1) * fct(j1 - m1) * fct(j1 + m1) *
                              fct(j2 - m2) * fct(j2 + m2)));
  double S = 0.0;
  for (int v = vmin; v <= vmax; ++v) {
    const double sgn = ((v + j2 + m2) & 1) ? -1.0 : 1.0;
    S += sgn * fct(j2 + j3 + m1 - v) * fct(j1 - m1 + v) /
         (fct(v) * fct(j3 - j1 + j2 - v) * fct(j3 + m3 - v) * fct(v + j1 - j2 - m3));
  }
  return C * S;
}

static void qmat(int l, cd* q) {               // (2l+1)x(2l+1) row-major
  const int d = 2 * l + 1;
  for (int i = 0; i < d * d; ++i) q[i] = cd(0, 0);
  const double s2 = 1.0 / std::sqrt(2.0);
  for (int m = -l; m < 0; ++m) {
    q[(l + m) * d + (l - m)] = cd(s2, 0);      // col l+|m|
    q[(l + m) * d + (l + m)] = cd(0, -s2);     // col l-|m|
  }
  q[l * d + l] = cd(1, 0);
  for (int m = 1; m <= l; ++m) {
    const double sg = (m & 1) ? -1.0 : 1.0;
    q[(l + m) * d + (l + m)] = cd(sg * s2, 0);
    q[(l + m) * d + (l - m)] = cd(0, sg * s2);
  }
  cd ph(1, 0);
  for (int i = 0; i < l; ++i) ph *= cd(0, -1); // (-1i)^l
  for (int i = 0; i < d * d; ++i) q[i] *= ph;
}

static void real_cg(int l1, int l2, int l3, float* outb) {
  const int d1 = 2 * l1 + 1, d2 = 2 * l2 + 1, d3 = 2 * l3 + 1;
  std::vector<double> C((size_t)d1 * d2 * d3, 0.0);
  for (int m1 = -l1; m1 <= l1; ++m1)
    for (int m2 = -l2; m2 <= l2; ++m2) {
      const int m3 = m1 + m2;
      if (std::abs(m3) <= l3)
        C[((size_t)(l1 + m1) * d2 + (l2 + m2)) * d3 + (l3 + m3)] = su2_cg(l1, m1, l2, m2, l3, m3);
    }
  std::vector<cd> Q1((size_t)d1 * d1), Q2((size_t)d2 * d2), Q3((size_t)d3 * d3);
  qmat(l1, Q1.data()); qmat(l2, Q2.data()); qmat(l3, Q3.data());
  // out[a][b][c] = sum_{i,k,n} Q1[i][a] Q2[k][b] conj(Q3[n][c]) C[i][k][n]
  for (int a = 0; a < d1; ++a)
    for (int b = 0; b < d2; ++b)
      for (int c = 0; c < d3; ++c) {
        cd s(0, 0);
        for (int i = 0; i < d1; ++i)
          for (int k = 0; k < d2; ++k)
            for (int n = 0; n < d3; ++n) {
              const double cv = C[((size_t)i * d2 + k) * d3 + n];
              if (cv != 0.0) s += Q1[i * d1 + a] * Q2[k * d2 + b] * std::conj(Q3[n * d3 + c]) * cv;
            }
        outb[((size_t)a * d2 + b) * d3 + c] = (float)s.real();
      }
}

static void build(float* t) {                  // pack order == device cg offsets
  static const int pairs[9][2] = {{1,1},{2,1},{3,1},{1,2},{2,2},{3,2},{1,3},{2,3},{3,3}};
  size_t ofs = 0;
  for (int p = 0; p < 9; ++p) {
    const int j = pairs[p][0], l = pairs[p][1];
    const int Jlo = std::abs(j - l), Jhi = std::min(j + l, 3);
    for (int J = Jlo; J <= Jhi; ++J) {
      real_cg(j, l, J, t + ofs);
      ofs += (size_t)(2 * j + 1) * (2 * l + 1) * (2 * J + 1);
    }
  }
}
} // namespace cgbuild

// persistent host buffer: graph-captured H2D memcpy must stay valid on replay
static float g_cg_host[3269];

extern "C" void kernel_launch(void* const* d_in, const int* in_sizes, int n_in,
                              void* d_out, int out_size, void* d_ws, size_t ws_size,
                              hipStream_t stream) {
  (void)in_sizes; (void)n_in; (void)out_size; (void)ws_size;
  const float* x0   = (const float*)d_in[0];
  const float* x1   = (const float*)d_in[1];
  const float* x2   = (const float*)d_in[2];
  const float* x3   = (const float*)d_in[3];
  const int*   pidx = (const int*)d_in[4];
  const float* kern = (const float*)d_in[5];

  cgbuild::build(g_cg_host);                                   // same values every call
  hipMemcpyAsync(d_ws, g_cg_host, sizeof(g_cg_host), hipMemcpyHostToDevice, stream);

  shconv_kernel<<<dim3(4 * 2048), dim3(TB), 0, stream>>>(
      x0, x1, x2, x3, pidx, kern, (const float*)d_ws, (float*)d_out);
}